// GraphSAGE_2456721293647
// MI455X (gfx1250) — compile-verified
//
#include <hip/hip_runtime.h>

// ---------------------------------------------------------------------------
// GraphSAGE 2-layer, fp32, MI455X (gfx1250).
// Aggregation (segment-mean) commutes with the linear projection, so layer 2
// projects h -> p = h @ W2_neigh (N x 64) BEFORE the edge scatter, cutting
// edge traffic from 150 floats/edge to 64 floats/edge. Dense GEMMs use
// V_WMMA_F32_16X16X4_F32 (full fp32 matrix pipe, matches reference precision).
// Weights are staged K-padded so the WMMA K-loop has zero branches.
// ---------------------------------------------------------------------------

#define NN      100000      // nodes (divisible by 16: 6250 tiles)
#define NE      1600000     // edges
#define INF     3
#define HID     150
#define HIDP    152         // K padded to 38*4 for WMMA K-loop
#define OUTF    64
#define NTILES  (NN / 16)   // 6250

typedef __attribute__((ext_vector_type(2))) float v2f;
typedef __attribute__((ext_vector_type(8))) float v8f;

// ---- 0) stage W2_self / W2_neigh into K-padded [HIDP][64] copies ---------
__global__ void sage_pad_w2(const float* __restrict__ W2s,
                            const float* __restrict__ W2n,
                            float* __restrict__ W2sp,   // [HIDP][64]
                            float* __restrict__ W2np) { // [HIDP][64]
    int idx = blockIdx.x * blockDim.x + threadIdx.x;    // HIDP*OUTF
    if (idx >= HIDP * OUTF) return;
    int k = idx / OUTF;
    W2sp[idx] = (k < HID) ? W2s[idx] : 0.0f;
    W2np[idx] = (k < HID) ? W2n[idx] : 0.0f;
}

// ---- 1) edge pass #1: degree + 3-wide aggregation of raw x ---------------
__global__ void sage_edge_agg1(const float* __restrict__ x,
                               const int* __restrict__ src,
                               const int* __restrict__ dst,
                               float* __restrict__ deg,
                               float* __restrict__ aggx) {
    int e = blockIdx.x * blockDim.x + threadIdx.x;
    if (e >= NE) return;
    int s = src[e], d = dst[e];
    atomicAdd(&deg[d], 1.0f);
#pragma unroll
    for (int c = 0; c < INF; ++c)
        atomicAdd(&aggx[d * INF + c], x[s * INF + c]);
}

// ---- 2) invdeg -----------------------------------------------------------
__global__ void sage_invdeg(const float* __restrict__ deg,
                            float* __restrict__ invdeg) {
    int n = blockIdx.x * blockDim.x + threadIdx.x;
    if (n < NN) invdeg[n] = 1.0f / fmaxf(deg[n], 1.0f);
}

// ---- 3) layer 1: h = relu(x@W1s + (aggx/deg)@W1n + b1), K-padded ---------
__global__ void sage_layer1(const float* __restrict__ x,
                            const float* __restrict__ aggx,
                            const float* __restrict__ invdeg,
                            const float* __restrict__ W1s,   // [3][150]
                            const float* __restrict__ W1n,   // [3][150]
                            const float* __restrict__ b1,    // [150]
                            float* __restrict__ h) {         // [NN][HIDP]
    int idx = blockIdx.x * blockDim.x + threadIdx.x;   // NN*HIDP = 15.2M
    if (idx >= NN * HIDP) return;
    int n = idx / HIDP;
    int j = idx - n * HIDP;
    if (j >= HID) { h[idx] = 0.0f; return; }           // zero the K padding
    float id  = invdeg[n];
    float acc = b1[j];
#pragma unroll
    for (int c = 0; c < INF; ++c) {
        acc += x[n * INF + c] * W1s[c * HID + j];
        acc += (aggx[n * INF + c] * id) * W1n[c * HID + j];
    }
    h[idx] = fmaxf(acc, 0.0f);
}

// ---- 4) WMMA GEMM: out = h@W2s + b2 ; p = h@W2n --------------------------
// One wave per 16-row tile of h. 8 waves / block. fp32 16x16x4 WMMA.
// A 16x4 f32 layout (ISA 7.12.2): lanes 0-15 carry K={k0,k0+1} in {v0,v1},
// lanes 16-31 carry K={k0+2,k0+3}. B 4x16 mirrored: N = lane&15 across lanes,
// K-pair selected by lane half. C/D 16x16 f32: col = lane&15, row = i + 8*half.
// Weights are pre-padded -> NO conditionals in the K loop.
__global__ void __launch_bounds__(256)
sage_gemm2(const float* __restrict__ h,    // [NN][HIDP]
           const float* __restrict__ W2sp, // [HIDP][64] padded
           const float* __restrict__ W2np, // [HIDP][64] padded
           const float* __restrict__ b2,   // [64]
           float* __restrict__ out,        // [NN][64]
           float* __restrict__ p) {        // [NN][64]
    const int wave = threadIdx.x >> 5;
    const int lane = threadIdx.x & 31;
    const int tile = blockIdx.x * 8 + wave;      // wave-uniform
    if (tile >= NTILES) return;                  // whole wave exits together

    const int half = lane >> 4;                  // 0 | 1
    const int r    = lane & 15;
    const float* __restrict__ arow = h + (size_t)tile * 16 * HIDP
                                       + (size_t)r * HIDP + 2 * half;
    const float* __restrict__ bs0 = W2sp + (size_t)(2 * half) * OUTF + r;
    const float* __restrict__ bn0 = W2np + (size_t)(2 * half) * OUTF + r;

    v8f accS[4], accN[4];
#pragma unroll
    for (int t = 0; t < 4; ++t) { accS[t] = (v8f)0.0f; accN[t] = (v8f)0.0f; }

    for (int k0 = 0; k0 < HIDP; k0 += 4) {
        v2f a;                                  // contiguous -> global_load_b64
        a.x = arow[k0 + 0];
        a.y = arow[k0 + 1];
#pragma unroll
        for (int t = 0; t < 4; ++t) {
            const int n0 = t * 16;
            v2f bs, bn;
            bs.x = bs0[(size_t)k0 * OUTF + n0];
            bs.y = bs0[(size_t)(k0 + 1) * OUTF + n0];
            bn.x = bn0[(size_t)k0 * OUTF + n0];
            bn.y = bn0[(size_t)(k0 + 1) * OUTF + n0];
            accS[t] = __builtin_amdgcn_wmma_f32_16x16x4_f32(
                false, a, false, bs, (short)0, accS[t], false, false);
            accN[t] = __builtin_amdgcn_wmma_f32_16x16x4_f32(
                false, a, false, bn, (short)0, accN[t], false, false);
        }
    }

#pragma unroll
    for (int t = 0; t < 4; ++t) {
        const int col = t * 16 + r;
        const float bias = b2[col];
#pragma unroll
        for (int i = 0; i < 8; ++i) {
            const size_t row = (size_t)tile * 16 + i + 8 * half;
            out[row * OUTF + col] = accS[t][i] + bias;
            p  [row * OUTF + col] = accN[t][i];
        }
    }
}

// ---- 5) edge pass #2: out[dst] += p[src] * invdeg[dst], 64-wide ----------
__global__ void sage_edge_agg2(const int* __restrict__ src,
                               const int* __restrict__ dst,
                               const float* __restrict__ p,
                               const float* __restrict__ invdeg,
                               float* __restrict__ out) {
    long idx = (long)blockIdx.x * blockDim.x + threadIdx.x;  // NE*16
    if (idx >= (long)NE * 16) return;
    int e  = (int)(idx >> 4);
    int c4 = ((int)idx & 15) * 4;                 // fast-varying -> coalesced
    int s = src[e], d = dst[e];
    float id = invdeg[d];
    const float4 v = *reinterpret_cast<const float4*>(p + (size_t)s * OUTF + c4);
    float* o = out + (size_t)d * OUTF + c4;
    atomicAdd(o + 0, v.x * id);
    atomicAdd(o + 1, v.y * id);
    atomicAdd(o + 2, v.z * id);
    atomicAdd(o + 3, v.w * id);
}

// ---------------------------------------------------------------------------
extern "C" void kernel_launch(void* const* d_in, const int* in_sizes, int n_in,
                              void* d_out, int out_size, void* d_ws, size_t ws_size,
                              hipStream_t stream) {
    const float* x   = (const float*)d_in[0];
    const int*   src = (const int*)  d_in[1];
    const int*   dst = (const int*)  d_in[2];
    const float* W1s = (const float*)d_in[3];
    const float* W1n = (const float*)d_in[4];
    const float* b1  = (const float*)d_in[5];
    const float* W2s = (const float*)d_in[6];
    const float* W2n = (const float*)d_in[7];
    const float* b2  = (const float*)d_in[8];
    float* out = (float*)d_out;

    // workspace layout (floats), 256B aligned slices
    char* ws = (char*)d_ws;
    auto align256 = [](size_t v) { return (v + 255) & ~(size_t)255; };
    size_t off = 0;
    float* deg    = (float*)(ws + off); off = align256(off + sizeof(float) * NN);
    float* invdeg = (float*)(ws + off); off = align256(off + sizeof(float) * NN);
    float* aggx   = (float*)(ws + off); off = align256(off + sizeof(float) * NN * INF);
    float* W2sp   = (float*)(ws + off); off = align256(off + sizeof(float) * HIDP * OUTF);
    float* W2np   = (float*)(ws + off); off = align256(off + sizeof(float) * HIDP * OUTF);
    float* h      = (float*)(ws + off); off = align256(off + sizeof(float) * (size_t)NN * HIDP);
    float* p      = (float*)(ws + off); off = align256(off + sizeof(float) * (size_t)NN * OUTF);
    (void)ws_size; (void)in_sizes; (void)n_in; (void)out_size;

    // zero the accumulated buffers (graph-capture-safe stream memsets)
    hipMemsetAsync(deg,  0, sizeof(float) * NN, stream);
    hipMemsetAsync(aggx, 0, sizeof(float) * NN * INF, stream);

    sage_pad_w2<<<(HIDP * OUTF + 255) / 256, 256, 0, stream>>>(W2s, W2n, W2sp, W2np);
    sage_edge_agg1<<<(NE + 255) / 256, 256, 0, stream>>>(x, src, dst, deg, aggx);
    sage_invdeg<<<(NN + 255) / 256, 256, 0, stream>>>(deg, invdeg);
    sage_layer1<<<(NN * HIDP + 255) / 256, 256, 0, stream>>>(
        x, aggx, invdeg, W1s, W1n, b1, h);
    sage_gemm2<<<(NTILES + 7) / 8, 256, 0, stream>>>(h, W2sp, W2np, b2, out, p);
    {
        long total = (long)NE * 16;
        sage_edge_agg2<<<(unsigned)((total + 255) / 256), 256, 0, stream>>>(
            src, dst, p, invdeg, out);
    }
}